// DensityPotential_20959440404551
// MI455X (gfx1250) — compile-verified
//
#include <hip/hip_runtime.h>
#include <math.h>

#define NBX   512
#define NBY   512
#define KW    5
#define NBINS (NBX * NBY)          // 262144
#define TARGET_AREA 0.9f

typedef __attribute__((ext_vector_type(2))) float v2f;
typedef __attribute__((ext_vector_type(8))) float v8f;

// ---------------------------------------------------------------------------
// Kernel 1: initialize workspace density map from initial_density_map, zero out
// ---------------------------------------------------------------------------
__global__ __launch_bounds__(256)
void dp_init_kernel(const float* __restrict__ init_map,
                    float* __restrict__ map,
                    float* __restrict__ out,
                    int nbins) {
    int i = blockIdx.x * blockDim.x + threadIdx.x;
    if (i < nbins) map[i] = init_map[i];
    if (i == 0)    out[0] = 0.0f;
}

// ---------------------------------------------------------------------------
// Kernel 2: per-node 5x5 separable potential scatter (global fp32 atomics,
// resolved in the 192MB L2 since the map is only 1MB)
// ---------------------------------------------------------------------------
__global__ __launch_bounds__(256)
void dp_scatter_kernel(const float* __restrict__ pos,
                       const float* __restrict__ nsx,
                       const float* __restrict__ nsy,
                       const float* __restrict__ axp,
                       const float* __restrict__ bxp,
                       const float* __restrict__ cxp,
                       const float* __restrict__ ayp,
                       const float* __restrict__ byp,
                       const float* __restrict__ cyp,
                       float* __restrict__ map,
                       int n) {
    int i = blockIdx.x * blockDim.x + threadIdx.x;
    if (i >= n) return;

    const float x  = pos[i];
    const float y  = pos[n + i];
    const float sx = nsx[i];
    const float sy = nsy[i];
    const float ax = axp[i], bx = bxp[i], cx = cxp[i];
    const float ay = ayp[i], by = byp[i], cy = cyp[i];

    // ---- X axis ----
    const float ctrx = x + 0.5f * sx;
    const float p1x  = 0.5f * sx + 1.0f;   // 0.5*size + BIN
    const float p2x  = 0.5f * sx + 2.0f;   // 0.5*size + 2*BIN
    int startx = (int)floorf(x - 2.0f);    // (x - 2*BIN - XL)/BIN, BIN=1, XL=0
    startx = min(max(startx, 0), NBX - KW);

    float px[KW];
    int   ixk[KW];
#pragma unroll
    for (int k = 0; k < KW; ++k) {
        const int   idx  = startx + k;
        const float bc   = (float)idx + 0.5f;          // bin_center_x[idx]
        const float dist = fabsf(ctrx - bc);
        float pot = 0.0f;
        if (dist < p1x) {
            pot = cx * (1.0f - ax * dist * dist);
        } else if (dist < p2x) {
            const float t = dist - p2x;
            pot = cx * bx * t * t;
        }
        px[k]  = pot;
        ixk[k] = idx;
    }

    // ---- Y axis ----
    const float ctry = y + 0.5f * sy;
    const float p1y  = 0.5f * sy + 1.0f;
    const float p2y  = 0.5f * sy + 2.0f;
    int starty = (int)floorf(y - 2.0f);
    starty = min(max(starty, 0), NBY - KW);

    float py[KW];
    int   iyk[KW];
#pragma unroll
    for (int k = 0; k < KW; ++k) {
        const int   idx  = starty + k;
        const float bc   = (float)idx + 0.5f;
        const float dist = fabsf(ctry - bc);
        float pot = 0.0f;
        if (dist < p1y) {
            pot = cy * (1.0f - ay * dist * dist);
        } else if (dist < p2y) {
            const float t = dist - p2y;
            pot = cy * by * t * t;
        }
        py[k]  = pot;
        iyk[k] = idx;
    }

    // ---- 25 scatter-adds (skip exact zeros to cut atomic traffic) ----
#pragma unroll
    for (int kx = 0; kx < KW; ++kx) {
        const int rowbase = ixk[kx] * NBY;
        const float pxv = px[kx];
        if (pxv == 0.0f) continue;
#pragma unroll
        for (int ky = 0; ky < KW; ++ky) {
            const float v = pxv * py[ky];
            if (v != 0.0f) {
                atomicAdd(&map[rowbase + iyk[ky]], v);
            }
        }
    }
}

// ---------------------------------------------------------------------------
// Kernel 3: cost = sum((map - target)^2), reduced on the matrix pipe.
// V_WMMA_F32_16X16X4_F32 with B = ones: D = A*1 + C accumulates 64 squared
// deltas per issue into 16 row-sums replicated across all 16 columns.
// Total accumulator sum = 16 * partial, so divide by 16 (exact).
// Layout: 32 blocks x 256 threads = 256 waves; each wave owns 1024 bins
// (16 WMMA iterations x 64 elements). 256*1024 == 262144 exactly, so EXEC is
// all-1s for every WMMA (required by the ISA).
// ---------------------------------------------------------------------------
__global__ __launch_bounds__(256)
void dp_reduce_kernel(const float* __restrict__ map,
                      float* __restrict__ out) {
    const int tid  = blockIdx.x * blockDim.x + threadIdx.x;
    const int wave = tid >> 5;
    const int lane = tid & 31;

    v8f c = {};
    v2f ones;
    ones.x = 1.0f;
    ones.y = 1.0f;

    const float* base = map + (size_t)wave * 1024 + (size_t)lane * 2;

#pragma unroll
    for (int it = 0; it < 16; ++it) {
        const float2 v = *(const float2*)(base + it * 64);
        const float d0 = v.x - TARGET_AREA;
        const float d1 = v.y - TARGET_AREA;
        v2f a;
        a.x = d0 * d0;
        a.y = d1 * d1;
        // (neg_a, A, neg_b, B, c_mod, C, reuse_a, reuse_b)
        c = __builtin_amdgcn_wmma_f32_16x16x4_f32(
                false, a, false, ones, (short)0, c, false, false);
    }

    // Sum this lane's 8 accumulator VGPRs, then wave32 shuffle reduction.
    float s = c[0] + c[1] + c[2] + c[3] + c[4] + c[5] + c[6] + c[7];
#pragma unroll
    for (int off = 16; off >= 1; off >>= 1) {
        s += __shfl_down(s, off, 32);
    }
    if (lane == 0) {
        atomicAdd(out, s * (1.0f / 16.0f));  // undo 16x column replication
    }
}

// ---------------------------------------------------------------------------
// Host-side launch
// ---------------------------------------------------------------------------
extern "C" void kernel_launch(void* const* d_in, const int* in_sizes, int n_in,
                              void* d_out, int out_size, void* d_ws, size_t ws_size,
                              hipStream_t stream) {
    const float* pos      = (const float*)d_in[0];
    const float* nsx      = (const float*)d_in[1];
    const float* nsy      = (const float*)d_in[2];
    const float* ax       = (const float*)d_in[3];
    const float* bx       = (const float*)d_in[4];
    const float* cx       = (const float*)d_in[5];
    const float* ay       = (const float*)d_in[6];
    const float* by       = (const float*)d_in[7];
    const float* cy       = (const float*)d_in[8];
    // d_in[9], d_in[10]: bin centers — analytic (XL + (i+0.5)*BIN), recomputed on device
    const float* init_map = (const float*)d_in[11];

    const int n = in_sizes[1];          // number of nodes (node_size_x length)
    float* map = (float*)d_ws;          // NBINS floats = 1 MB scratch density map
    float* out = (float*)d_out;         // scalar cost

    dp_init_kernel<<<(NBINS + 255) / 256, 256, 0, stream>>>(init_map, map, out, NBINS);

    dp_scatter_kernel<<<(n + 255) / 256, 256, 0, stream>>>(
        pos, nsx, nsy, ax, bx, cx, ay, by, cy, map, n);

    // 32 blocks * 8 waves * 16 iters * 64 elems = 262144 bins exactly
    dp_reduce_kernel<<<32, 256, 0, stream>>>(map, out);
}